// MixedDilatedAttentionDeprecated_53429393162515
// MI455X (gfx1250) — compile-verified
//
#include <hip/hip_runtime.h>

typedef float v2f __attribute__((ext_vector_type(2)));
typedef float v8f __attribute__((ext_vector_type(8)));
typedef unsigned int u32x4 __attribute__((ext_vector_type(4)));
typedef int i32x8 __attribute__((ext_vector_type(8)));
typedef int i32x4 __attribute__((ext_vector_type(4)));

#define NH    16
#define HD    64
#define TSEQ  2048
#define BATCH 8
#define CIN   1024
#define NOUT  3072

#if defined(__has_builtin)
#if __has_builtin(__builtin_amdgcn_tensor_load_to_lds)
#define HAVE_TDM 1
#else
#define HAVE_TDM 0
#endif
#else
#define HAVE_TDM 0
#endif

__device__ __forceinline__ v8f vzero8() {
  v8f z;
  for (int i = 0; i < 8; i++) z[i] = 0.0f;
  return z;
}

// D = A(16x4 f32) * B(4x16 f32) + C(16x16 f32), full-wave32 op.
__device__ __forceinline__ v8f wmma_f32(v2f a, v2f b, v8f c) {
  return __builtin_amdgcn_wmma_f32_16x16x4_f32(
      false, a, false, b, (short)0, c, false, false);
}

// Async global->LDS copies (ASYNCcnt). lds_off = byte offset in LDS
// (low 32 bits of the generic pointer to __shared__).
__device__ __forceinline__ void async_load_b128(unsigned lds_off, const void* g) {
  asm volatile("global_load_async_to_lds_b128 %0, %1, off"
               :: "v"(lds_off), "v"((unsigned long long)(uintptr_t)g)
               : "memory");
}
__device__ __forceinline__ void async_load_b32(unsigned lds_off, const void* g) {
  asm volatile("global_load_async_to_lds_b32 %0, %1, off"
               :: "v"(lds_off), "v"((unsigned long long)(uintptr_t)g)
               : "memory");
}
__device__ __forceinline__ void wait_async0() {
  asm volatile("s_wait_asynccnt 0x0" ::: "memory");
}
__device__ __forceinline__ void wait_tensor0() {
#if defined(__has_builtin)
#if __has_builtin(__builtin_amdgcn_s_wait_tensorcnt)
  __builtin_amdgcn_s_wait_tensorcnt(0);
#else
  asm volatile("s_wait_tensorcnt 0x0" ::: "memory");
#endif
#else
  asm volatile("s_wait_tensorcnt 0x0" ::: "memory");
#endif
}

// ---------------------------------------------------------------------------
// QKV GEMM: qkv[m, n] = x[m, :] . w[:, n] + bias[n], written head-major
// [3][B][NH][T][HD]. 64x64 block tile, 4 waves of 32x32, fp32 WMMA 16x16x4.
// Double-buffered K chunks of 16: A staged via global_load_async_to_lds_b128,
// B staged via TDM tensor_load_to_lds (16x64 tile, padded to 68-dword pitch).
// ---------------------------------------------------------------------------
__global__ __launch_bounds__(128) void qkv_gemm_kernel(
    const float* __restrict__ X, const float* __restrict__ W,
    const float* __restrict__ Bias, float* __restrict__ QKV) {
  constexpr int PA = 20;   // A pitch (dwords): 80B rows, 16B aligned
  constexpr int PBR = 68;  // B raw pitch (dwords): 64 data + 4 TDM pad
  __shared__ float As[2][64 * PA];
  __shared__ float Bs[2][16 * PBR];

  const int n0 = blockIdx.x * 64;
  const int m0 = blockIdx.y * 64;
  const int tid = threadIdx.x;
  const int lane = tid & 31;
  const int wave = tid >> 5;
  const int wm = wave >> 1, wn = wave & 1;
  const int l16 = lane & 15;
  const int lhi = lane >> 4;

  v8f acc[2][2];
  for (int i = 0; i < 2; i++)
    for (int j = 0; j < 2; j++) acc[i][j] = vzero8();

  auto stage = [&](int kc, int buf) {
    // A tile 64x16: 2 async b128 per thread.
#pragma unroll
    for (int f = tid; f < 256; f += 128) {
      const int row = f >> 2, cg = (f & 3) * 4;
      async_load_b128((unsigned)(uintptr_t)&As[buf][row * PA + cg],
                      X + (size_t)(m0 + row) * CIN + kc + cg);
    }
#if HAVE_TDM
    if (wave == 0) {
      // Tensor DMA descriptor: 16x64 f32 tile of w at (kc, n0), row stride 3072,
      // LDS-padded 4 dwords per 64-dword row.
      const unsigned lds = (unsigned)(uintptr_t)&Bs[buf][0];
      const unsigned long long ga =
          (unsigned long long)(uintptr_t)(W + (size_t)kc * NOUT + n0);
      u32x4 g0;
      g0[0] = 1u;                                   // count=1 valid descriptor
      g0[1] = lds;                                  // lds_addr
      g0[2] = (unsigned)(ga & 0xFFFFFFFFu);         // global_addr[31:0]
      g0[3] = (unsigned)((ga >> 32) & 0x1FFFFFFu)   // global_addr[56:32]
              | 0x80000000u;                        // type=2 ("image")
      i32x8 g1;
      g1[0] = (int)((2u << 16)      // data_size = 4B
                    | (1u << 20)    // pad_enable
                    | (5u << 22)    // pad_interval: 64 dwords
                    | (3u << 25));  // pad_amount: 4 dwords
      g1[1] = (int)((unsigned)NOUT << 16);    // tensor_dim0[15:0]
      g1[2] = (int)((NOUT >> 16) | (1024u << 16)); // dim0[31:16] | dim1[15:0]
      g1[3] = (int)(64u << 16);               // dim1[31:16]=0 | tile_dim0=64
      g1[4] = 16;                             // tile_dim1=16, tile_dim2=0
      g1[5] = NOUT;                           // tensor_dim0_stride[31:0]
      g1[6] = 0;                              // stride0[47:32] | stride1[15:0]
      g1[7] = 0;
      i32x4 z4 = {0, 0, 0, 0};
      i32x8 z8 = {0, 0, 0, 0, 0, 0, 0, 0};
      __builtin_amdgcn_tensor_load_to_lds(g0, g1, z4, z4, z8, 0);
    }
#else
    // Fallback: manual B staging into the same raw [k][PBR] layout.
#pragma unroll
    for (int f = tid; f < 256; f += 128) {
      const int kr = f >> 4, ng = (f & 15) * 4;
      float4 v = *(const float4*)(W + (size_t)(kc + kr) * NOUT + n0 + ng);
      *(float4*)&Bs[buf][kr * PBR + ng] = v;
    }
#endif
  };

  stage(0, 0);
  wait_async0();
  wait_tensor0();
  __syncthreads();

  for (int kc = 0; kc < CIN; kc += 16) {
    const int cur = (kc >> 4) & 1;
    if (kc + 16 < CIN) stage(kc + 16, cur ^ 1);  // overlap DMA with WMMA

#pragma unroll
    for (int kk = 0; kk < 16; kk += 4) {
      v2f a[2], bb[2];
#pragma unroll
      for (int mt = 0; mt < 2; mt++)
        a[mt] = *(const v2f*)&As[cur][(32 * wm + 16 * mt + l16) * PA + kk + 2 * lhi];
#pragma unroll
      for (int nt = 0; nt < 2; nt++) {
        const float* bp = &Bs[cur][(kk + 2 * lhi) * PBR + 32 * wn + 16 * nt + l16];
        v2f b = {bp[0], bp[PBR]};
        bb[nt] = b;
      }
#pragma unroll
      for (int mt = 0; mt < 2; mt++)
#pragma unroll
        for (int nt = 0; nt < 2; nt++)
          acc[mt][nt] = wmma_f32(a[mt], bb[nt], acc[mt][nt]);
    }

    wait_async0();
    wait_tensor0();
    __syncthreads();
  }

  // Epilogue: bias + scatter to [which][b][h][t][ch].
#pragma unroll
  for (int mt = 0; mt < 2; mt++) {
#pragma unroll
    for (int nt = 0; nt < 2; nt++) {
      const int ncol = n0 + 32 * wn + 16 * nt + l16;
      const float bv = Bias[ncol];
      const int which = ncol >> 10;
      const int rem = ncol & 1023;
      const int h = rem >> 6, ch = rem & 63;
#pragma unroll
      for (int r = 0; r < 8; r++) {
        const int m = m0 + 32 * wm + 16 * mt + r + 8 * lhi;
        const int bI = m >> 11, t = m & (TSEQ - 1);
        QKV[((((size_t)which * BATCH + bI) * NH + h) * TSEQ + t) * HD + ch] =
            acc[mt][nt][r] + bv;
      }
    }
  }
}

// ---------------------------------------------------------------------------
// One (W, d) dilated-attention pair. WD = W/d rows, CD = 64/d channels.
// Block = one (batch, head, window); WD/16 waves, each owning a 16-row strip.
// ---------------------------------------------------------------------------
template <int WD, int CD, int D>
__global__ __launch_bounds__(256) void attn_pair_kernel(
    const float* __restrict__ Q, const float* __restrict__ K,
    const float* __restrict__ V, float* __restrict__ Y, float coef) {
  constexpr int WIN = WD * D;
  constexpr int WAVES = WD / 16;
  constexpr int CDP = (CD < 16) ? 16 : CD;  // pad V cols for 16-wide WMMA N
  constexpr int PQ = CD + 2;
  constexpr int PV = WD + 2;
  constexpr int PS = WD + 2;
  extern __shared__ float smem[];
  float* Qs = smem;               // [WD][PQ]
  float* Ks = Qs + WD * PQ;       // [WD][PQ]
  float* Vt = Ks + WD * PQ;       // [CDP][PV]  (V transposed, zero-padded)
  float* Ss = Vt + CDP * PV;      // [WD][PS]   (scores, then probabilities)

  const int tid = threadIdx.x;
  const int nthr = WD * 2;
  const int lane = tid & 31;
  const int wave = tid >> 5;
  const int l16 = lane & 15;
  const int lhi = lane >> 4;

  const int w0 = blockIdx.x * WIN;
  const int bh = blockIdx.y;  // b*NH + h
  const size_t base = ((size_t)bh * TSEQ + w0) * HD;
  const float* qg = Q + base;
  const float* kg = K + base;
  const float* vg = V + base;

  // Strided gathers straight into LDS (async, no VGPR round-trip).
  for (int idx = tid; idx < WD * CD; idx += nthr) {
    const int i = idx / CD, j = idx % CD;
    async_load_b32((unsigned)(uintptr_t)&Qs[i * PQ + j],
                   qg + (size_t)i * D * HD + j * D);
    async_load_b32((unsigned)(uintptr_t)&Ks[i * PQ + j],
                   kg + (size_t)i * D * HD + j * D);
  }
  for (int idx = tid; idx < CD * WD; idx += nthr) {
    const int j = idx / WD, i = idx % WD;
    async_load_b32((unsigned)(uintptr_t)&Vt[j * PV + i],
                   vg + (size_t)i * D * HD + j * D);
  }
  if (CDP > CD) {  // zero-pad V^T rows [CD, CDP)
    for (int idx = tid; idx < (CDP - CD) * PV; idx += nthr)
      Vt[CD * PV + idx] = 0.0f;
  }
  wait_async0();
  __syncthreads();

  // Scores S = Qc . Kc^T * scale; wave computes its 16-row strip x all cols.
  v8f sacc[WAVES];
  for (int jt = 0; jt < WAVES; jt++) sacc[jt] = vzero8();
#pragma unroll
  for (int kk = 0; kk < CD; kk += 4) {
    v2f a = *(const v2f*)&Qs[(16 * wave + l16) * PQ + kk + 2 * lhi];
#pragma unroll
    for (int jt = 0; jt < WAVES; jt++) {
      v2f b = *(const v2f*)&Ks[(16 * jt + l16) * PQ + kk + 2 * lhi];
      sacc[jt] = wmma_f32(a, b, sacc[jt]);
    }
  }
  const float scale = 0.125f;  // 1/sqrt(64)
#pragma unroll
  for (int jt = 0; jt < WAVES; jt++) {
    const int jc = 16 * jt + l16;
#pragma unroll
    for (int r = 0; r < 8; r++) {
      const int i = 16 * wave + r + 8 * lhi;
      const float v = sacc[jt][r] * scale;
      Ss[i * PS + jc] = (jc <= i) ? v : -__builtin_huge_valf();
    }
  }
  __syncthreads();

  // Softmax: one wave per row (looped), shfl_xor reductions.
  for (int row = wave; row < WD; row += WAVES) {
    float vals[WD / 32];
    float m = -__builtin_huge_valf();
#pragma unroll
    for (int c = 0; c < WD / 32; c++) {
      vals[c] = Ss[row * PS + lane + 32 * c];
      m = fmaxf(m, vals[c]);
    }
#pragma unroll
    for (int o = 16; o > 0; o >>= 1) m = fmaxf(m, __shfl_xor(m, o, 32));
    float s = 0.0f;
#pragma unroll
    for (int c = 0; c < WD / 32; c++) {
      vals[c] = __expf(vals[c] - m);
      s += vals[c];
    }
#pragma unroll
    for (int o = 16; o > 0; o >>= 1) s += __shfl_xor(s, o, 32);
    const float inv = 1.0f / s;
#pragma unroll
    for (int c = 0; c < WD / 32; c++) Ss[row * PS + lane + 32 * c] = vals[c] * inv;
  }
  __syncthreads();

  // O = P . Vc  (K dim = WD).
  constexpr int NT = CDP / 16;
  v8f oacc[NT];
  for (int nt = 0; nt < NT; nt++) oacc[nt] = vzero8();
#pragma unroll 4
  for (int kk = 0; kk < WD; kk += 4) {
    v2f a = *(const v2f*)&Ss[(16 * wave + l16) * PS + kk + 2 * lhi];
#pragma unroll
    for (int nt = 0; nt < NT; nt++) {
      v2f b = *(const v2f*)&Vt[(16 * nt + l16) * PV + kk + 2 * lhi];
      oacc[nt] = wmma_f32(a, b, oacc[nt]);
    }
  }

  // Scatter coef * O to y[b][t = w0+i*D][h*64 + n*D].
  const int bI = bh >> 4, h = bh & 15;
#pragma unroll
  for (int nt = 0; nt < NT; nt++) {
    const int n = 16 * nt + l16;
    if (n < CD) {
      const int ch = n * D;
#pragma unroll
      for (int r = 0; r < 8; r++) {
        const int i = 16 * wave + r + 8 * lhi;
        const int t = w0 + i * D;
        float* yp = Y + ((size_t)bI * TSEQ + t) * CIN + h * HD + ch;
        const float val = oacc[nt][r] * coef;
        if (D == 1) *yp = val;       // dense pair: covers every output elem
        else atomicAdd(yp, val);     // sparse pairs accumulate
      }
    }
  }
}

template <int WD, int CD>
constexpr size_t attn_smem_bytes() {
  constexpr int CDP = (CD < 16) ? 16 : CD;
  return ((size_t)WD * (CD + 2) * 2 + (size_t)CDP * (WD + 2) +
          (size_t)WD * (WD + 2)) * sizeof(float);
}

extern "C" void kernel_launch(void* const* d_in, const int* in_sizes, int n_in,
                              void* d_out, int out_size, void* d_ws, size_t ws_size,
                              hipStream_t stream) {
  (void)in_sizes; (void)n_in; (void)out_size; (void)ws_size;
  const float* x = (const float*)d_in[0];
  const float* w = (const float*)d_in[1];
  const float* bias = (const float*)d_in[2];
  float* y = (float*)d_out;

  float* qkv = (float*)d_ws;  // [3][B][NH][T][HD] = 192 MB
  const size_t sec = (size_t)BATCH * NH * TSEQ * HD;
  float* q = qkv;
  float* k = qkv + sec;
  float* v = qkv + 2 * sec;

  qkv_gemm_kernel<<<dim3(NOUT / 64, (BATCH * TSEQ) / 64), 128, 0, stream>>>(
      x, w, bias, qkv);

  const int BH = BATCH * NH;
  // Pair weights: pd = B*nh*nW*Wd exactly (each causal softmax row sums to 1)
  // => coef = (16/d)/31.
  attn_pair_kernel<64, 64, 1><<<dim3(32, BH), 128, attn_smem_bytes<64, 64>(), stream>>>(
      q, k, v, y, 16.0f / 31.0f);
  attn_pair_kernel<64, 32, 2><<<dim3(16, BH), 128, attn_smem_bytes<64, 32>(), stream>>>(
      q, k, v, y, 8.0f / 31.0f);
  attn_pair_kernel<128, 16, 4><<<dim3(4, BH), 256, attn_smem_bytes<128, 16>(), stream>>>(
      q, k, v, y, 4.0f / 31.0f);
  attn_pair_kernel<128, 8, 8><<<dim3(2, BH), 256, attn_smem_bytes<128, 8>(), stream>>>(
      q, k, v, y, 2.0f / 31.0f);
  attn_pair_kernel<128, 4, 16><<<dim3(1, BH), 256, attn_smem_bytes<128, 4>(), stream>>>(
      q, k, v, y, 1.0f / 31.0f);
}